// GNNModel_10788957847586
// MI455X (gfx1250) — compile-verified
//
#include <hip/hip_runtime.h>
#include <hip/hip_bf16.h>

// ---------------------------------------------------------------------------
// GNN: 2x TransformerConv (heads=1) + linear decoder, f32 end-to-end.
//  - Dense GEMMs: V_WMMA_F32_16X16X4_F32, one wave per 16-row strip computing
//    all N/16 column tiles (A frag reused across 6 WMMAs), W staged in LDS
//    via TDM tensor_load_to_lds (6-arg toolchain form).
//  - Edge softmax/aggregation: gather/scatter kernels; true bottleneck
//    (~2-3 GB irregular traffic -> HBM/L2 bound at 23.3 TB/s).
// ---------------------------------------------------------------------------

#define N_NODES 50000
#define N_EDGES 800000
#define D_IN    128
#define D_HID   96
#define D_OUT   32

typedef float v2f __attribute__((ext_vector_type(2)));
typedef float v8f __attribute__((ext_vector_type(8)));
typedef unsigned int u32;
typedef u32 v4u __attribute__((ext_vector_type(4)));
typedef int  v4i __attribute__((ext_vector_type(4)));
typedef int  v8i __attribute__((ext_vector_type(8)));

#if __has_builtin(__builtin_amdgcn_tensor_load_to_lds)
#define USE_TDM 1
#endif

#ifdef USE_TDM
// Issue a TDM load of a 2-D row-major f32 tile (rows x cols == whole tensor)
// from global memory into LDS. D# bitfields per CDNA5 ISA §8.3/8.4:
//   group0: [1:0]=count=1, [63:32]=lds_addr, [120:64]=global_addr, [127:126]=type=2
//   group1: [17:16]=data_size(2->4B), [79:48]=tensor_dim0, [111:80]=tensor_dim1,
//           [127:112]=tile_dim0, [143:128]=tile_dim1, [207:160]=dim0_stride
// Groups 2/3 (tensor_dim2..4 etc.) are zero for a 2-D tile.
__device__ __forceinline__ void tdm_load_2d_f32(const float* gptr, u32 lds_off,
                                                u32 cols, u32 rows)
{
    unsigned long long ga = (unsigned long long)(uintptr_t)gptr;
    v4u g0;
    g0[0] = 1u;                                          // count=1, user mode
    g0[1] = lds_off;                                     // lds_addr (bytes)
    g0[2] = (u32)(ga & 0xFFFFFFFFull);                   // global_addr[31:0]
    g0[3] = (u32)((ga >> 32) & 0x1FFFFFFull) | (2u << 30); // addr[56:32] | type=2
    v8i g1;
    g1[0] = (int)(2u << 16);                             // data_size=4B, wg_mask=0
    g1[1] = (int)((cols & 0xFFFFu) << 16);               // tensor_dim0 lo16 @ [63:48]
    g1[2] = (int)(((cols >> 16) & 0xFFFFu) | ((rows & 0xFFFFu) << 16)); // dim0 hi | dim1 lo
    g1[3] = (int)(((rows >> 16) & 0xFFFFu) | ((cols & 0xFFFFu) << 16)); // dim1 hi | tile_dim0
    g1[4] = (int)(rows & 0xFFFFu);                       // tile_dim1 (tile_dim2=0)
    g1[5] = (int)cols;                                   // tensor_dim0_stride lo32
    g1[6] = 0;                                           // stride0 hi16 | stride1 lo16
    g1[7] = 0;
    v4i z4 = {0, 0, 0, 0};
    v8i z8 = {0, 0, 0, 0, 0, 0, 0, 0};
    // 6-arg toolchain form: (g0, g1, g2, g3, extra, cpol)
    __builtin_amdgcn_tensor_load_to_lds(g0, g1, z4, z4, z8, 0);
}
#endif

// ---------------------------------------------------------------------------
// WMMA f32 GEMM: out[M,N] = act(A[M,K]) @ W[K,N] + bias.
// Block = 8 waves; each wave owns one 16-row strip and computes all N/16
// column tiles. W (K*N*4 <= 48KB) staged in LDS once per block.
//
// V_WMMA_F32_16X16X4_F32 operand layouts (ISA 7.12.2, wave32):
//   A (16x4): lanes 0-15: row M=lane,    VGPR0/1 = K=kb+0,kb+1
//             lanes 16-31: row M=lane-16, VGPR0/1 = K=kb+2,kb+3
//   B (4x16): mirrored per-lane K split, col N = lane&15
//   C/D (16x16): VGPR r: lanes 0-15 -> M=r, N=lane; lanes 16-31 -> M=r+8
// ---------------------------------------------------------------------------
template <int K, int N, bool RELU>
__global__ __launch_bounds__(256) void wmma_gemm_f32(const float* __restrict__ A,
                                                     const float* __restrict__ W,
                                                     const float* __restrict__ bias,
                                                     float* __restrict__ out, int M)
{
    constexpr int NT = N / 16;
    __shared__ float sW[K * N];

    const int lane = threadIdx.x;                        // 0..31 (wave32)
    const int wave = threadIdx.y;                        // 0..7

#ifdef USE_TDM
    if (wave == 0) {
        tdm_load_2d_f32(W, (u32)(uintptr_t)(&sW[0]), N, K);
        __builtin_amdgcn_s_wait_tensorcnt(0);            // TDM done before barrier
    }
#else
    for (int i = lane + 32 * wave; i < K * N; i += 256) sW[i] = W[i];
#endif
    __syncthreads();

    const int tileM = blockIdx.x * 8 + wave;
    const int m0 = tileM << 4;
    if (m0 >= M) return;                                 // whole-wave exit, EXEC all-1s

    const int mrow  = lane & 15;
    const int ncol  = lane & 15;
    const int khalf = (lane >> 4) << 1;                  // 0 | 2

    const float* __restrict__ arow = A + (size_t)(m0 + mrow) * K;

    v8f c[NT];
#pragma unroll
    for (int t = 0; t < NT; ++t) c[t] = (v8f){};

#pragma unroll
    for (int kb = 0; kb < K; kb += 4) {
        float a0 = arow[kb + khalf];
        float a1 = arow[kb + khalf + 1];
        if (RELU) { a0 = fmaxf(a0, 0.0f); a1 = fmaxf(a1, 0.0f); }
        v2f a; a.x = a0; a.y = a1;
#pragma unroll
        for (int t = 0; t < NT; ++t) {
            v2f b;
            b.x = sW[(kb + khalf)     * N + t * 16 + ncol];
            b.y = sW[(kb + khalf + 1) * N + t * 16 + ncol];
            // 8 args: (neg_a, A, neg_b, B, c_mod, C, reuse_a, reuse_b)
            c[t] = __builtin_amdgcn_wmma_f32_16x16x4_f32(false, a, false, b,
                                                         (short)0, c[t], false, false);
        }
    }

    const int rbase = (lane >> 4) << 3;                  // 0 | 8
#pragma unroll
    for (int t = 0; t < NT; ++t) {
        const float bv = bias[t * 16 + ncol];
#pragma unroll
        for (int r = 0; r < 8; ++r)
            out[(size_t)(m0 + rbase + r) * N + t * 16 + ncol] = c[t][r] + bv;
    }
}

// ---------------------------------------------------------------------------
// Edge-stage kernels (segment softmax attention)
// ---------------------------------------------------------------------------
__global__ void init_ms_kernel(float* __restrict__ m, float* __restrict__ s, int n)
{
    int i = blockIdx.x * blockDim.x + threadIdx.x;
    if (i < n) { m[i] = -__builtin_inff(); s[i] = 0.0f; }
}

__device__ __forceinline__ void atomicMaxF(float* addr, float v)
{
    if (v >= 0.0f) atomicMax((int*)addr, __float_as_int(v));
    else           atomicMin((unsigned int*)addr, __float_as_uint(v));
}

// alpha[e] = scale * dot(q[dst[e]], k[src[e]]); segment max into m[dst].
__global__ void edge_alpha_kernel(const float* __restrict__ q,
                                  const float* __restrict__ k,
                                  const int* __restrict__ src,
                                  const int* __restrict__ dst,
                                  float* __restrict__ alpha,
                                  float* __restrict__ m,
                                  int E, float scale)
{
    int tid = blockIdx.x * blockDim.x + threadIdx.x;
    int e = tid >> 3;
    int l = tid & 7;
    if (e >= E) return;

    int sn = src[e];
    int dn = dst[e];
    const float4* __restrict__ qr = (const float4*)(q + (size_t)dn * D_HID);
    const float4* __restrict__ kr = (const float4*)(k + (size_t)sn * D_HID);

    float acc = 0.0f;
#pragma unroll
    for (int j = 0; j < 3; ++j) {                        // 24 float4 / 8 lanes
        float4 qa = qr[l + j * 8];
        float4 ka = kr[l + j * 8];
        acc += qa.x * ka.x + qa.y * ka.y + qa.z * ka.z + qa.w * ka.w;
    }
    acc += __shfl_xor(acc, 1, 8);
    acc += __shfl_xor(acc, 2, 8);
    acc += __shfl_xor(acc, 4, 8);

    if (l == 0) {
        float a = acc * scale;
        alpha[e] = a;
        atomicMaxF(&m[dn], a);
    }
}

// alpha[e] <- exp(alpha[e] - m[dst]); segment sum into s[dst].
__global__ void edge_expsum_kernel(const int* __restrict__ dst,
                                   float* __restrict__ alpha,
                                   const float* __restrict__ m,
                                   float* __restrict__ s, int E)
{
    int e = blockIdx.x * blockDim.x + threadIdx.x;
    if (e >= E) return;
    int dn = dst[e];
    float ev = __expf(alpha[e] - m[dn]);
    alpha[e] = ev;
    atomicAdd(&s[dn], ev);
}

// out[dst] += (alpha[e]/s[dst]) * v[src]; out pre-initialized with skip branch.
__global__ void edge_aggregate_kernel(const float* __restrict__ v,
                                      const int* __restrict__ src,
                                      const int* __restrict__ dst,
                                      const float* __restrict__ alpha,
                                      const float* __restrict__ s,
                                      float* __restrict__ out, int E)
{
    int tid = blockIdx.x * blockDim.x + threadIdx.x;
    int e = tid >> 5;
    int lane = tid & 31;
    if (e >= E) return;

    int sn = src[e];
    int dn = dst[e];
    float w = alpha[e] / (s[dn] + 1e-16f);
    const float* __restrict__ vr = v + (size_t)sn * D_HID;
    float* __restrict__ orow = out + (size_t)dn * D_HID;
#pragma unroll
    for (int j = 0; j < 3; ++j) {
        int f = lane + j * 32;
        atomicAdd(&orow[f], w * vr[f]);
    }
}

// ---------------------------------------------------------------------------
// Orchestration
// ---------------------------------------------------------------------------
extern "C" void kernel_launch(void* const* d_in, const int* in_sizes, int n_in,
                              void* d_out, int out_size, void* d_ws, size_t ws_size,
                              hipStream_t stream)
{
    const float* x    = (const float*)d_in[0];
    const int*   ei   = (const int*)d_in[1];       // [2, E] (JAX default int32)
    const float* Wq1  = (const float*)d_in[2];   const float* bq1 = (const float*)d_in[3];
    const float* Wk1  = (const float*)d_in[4];   const float* bk1 = (const float*)d_in[5];
    const float* Wv1  = (const float*)d_in[6];   const float* bv1 = (const float*)d_in[7];
    const float* Ws1  = (const float*)d_in[8];   const float* bs1 = (const float*)d_in[9];
    const float* Wq2  = (const float*)d_in[10];  const float* bq2 = (const float*)d_in[11];
    const float* Wk2  = (const float*)d_in[12];  const float* bk2 = (const float*)d_in[13];
    const float* Wv2  = (const float*)d_in[14];  const float* bv2 = (const float*)d_in[15];
    const float* Ws2  = (const float*)d_in[16];  const float* bs2 = (const float*)d_in[17];
    const float* Wd   = (const float*)d_in[18];  const float* bd  = (const float*)d_in[19];

    const int* src = ei;
    const int* dst = ei + N_EDGES;

    // Workspace layout (floats): ~99.6 MB total
    float* ws    = (float*)d_ws;
    const size_t NH = (size_t)N_NODES * D_HID;          // 4.8M floats
    float* q     = ws;
    float* k     = ws + NH;
    float* v     = ws + 2 * NH;
    float* h1    = ws + 3 * NH;
    float* h2    = ws + 4 * NH;
    float* alpha = ws + 5 * NH;                          // N_EDGES floats
    float* m     = ws + 5 * NH + N_EDGES;
    float* s     = ws + 5 * NH + N_EDGES + N_NODES;

    const float scale = 1.0f / sqrtf((float)D_HID);

    const int tilesM = N_NODES / 16;                     // 3125 exactly
    dim3 gblk(32, 8);                                    // 8 waves / block
    dim3 ggrid((tilesM + 7) / 8, 1);

    const int TA = (N_EDGES * 8 + 255) / 256;
    const int TE = (N_EDGES + 255) / 256;
    const int TG = (N_EDGES * 32 + 255) / 256;
    const int TN = (N_NODES + 255) / 256;

    // ---- Layer 1: q/k/v and skip-initialized h1 from x (K=128) ----
    wmma_gemm_f32<D_IN, D_HID, false><<<ggrid, gblk, 0, stream>>>(x, Wq1, bq1, q,  N_NODES);
    wmma_gemm_f32<D_IN, D_HID, false><<<ggrid, gblk, 0, stream>>>(x, Wk1, bk1, k,  N_NODES);
    wmma_gemm_f32<D_IN, D_HID, false><<<ggrid, gblk, 0, stream>>>(x, Wv1, bv1, v,  N_NODES);
    wmma_gemm_f32<D_IN, D_HID, false><<<ggrid, gblk, 0, stream>>>(x, Ws1, bs1, h1, N_NODES);

    init_ms_kernel<<<TN, 256, 0, stream>>>(m, s, N_NODES);
    edge_alpha_kernel    <<<TA, 256, 0, stream>>>(q, k, src, dst, alpha, m, N_EDGES, scale);
    edge_expsum_kernel   <<<TE, 256, 0, stream>>>(dst, alpha, m, s, N_EDGES);
    edge_aggregate_kernel<<<TG, 256, 0, stream>>>(v, src, dst, alpha, s, h1, N_EDGES);

    // ---- Layer 2: inputs are relu(h1), fused into GEMM A-loads (K=96) ----
    wmma_gemm_f32<D_HID, D_HID, true><<<ggrid, gblk, 0, stream>>>(h1, Wq2, bq2, q,  N_NODES);
    wmma_gemm_f32<D_HID, D_HID, true><<<ggrid, gblk, 0, stream>>>(h1, Wk2, bk2, k,  N_NODES);
    wmma_gemm_f32<D_HID, D_HID, true><<<ggrid, gblk, 0, stream>>>(h1, Wv2, bv2, v,  N_NODES);
    wmma_gemm_f32<D_HID, D_HID, true><<<ggrid, gblk, 0, stream>>>(h1, Ws2, bs2, h2, N_NODES);

    init_ms_kernel<<<TN, 256, 0, stream>>>(m, s, N_NODES);
    edge_alpha_kernel    <<<TA, 256, 0, stream>>>(q, k, src, dst, alpha, m, N_EDGES, scale);
    edge_expsum_kernel   <<<TE, 256, 0, stream>>>(dst, alpha, m, s, N_EDGES);
    edge_aggregate_kernel<<<TG, 256, 0, stream>>>(v, src, dst, alpha, s, h2, N_EDGES);

    // ---- Decoder: d_out = h2 @ Wd + bd (K=96, N=32) ----
    wmma_gemm_f32<D_HID, D_OUT, false><<<ggrid, gblk, 0, stream>>>(h2, Wd, bd,
                                                                   (float*)d_out, N_NODES);
}